// GCNIILayer_21852793602415
// MI455X (gfx1250) — compile-verified
//
#include <hip/hip_runtime.h>

#define NN 50000
#define NE 800000
#define DD 64
#define ALPHA 0.1f
#define BETA 0.5f
#define ROW_TILES (NN / 16)   // 3125 exactly
#define WAVES_PER_BLOCK 8
// K-pair row stride in floats: 128 data + 32 pad -> stride mod 64 banks == 32,
// so the two EXEC halves of a wave hit disjoint bank sets on ds_load_b64.
#define KP 160

typedef float v2f __attribute__((ext_vector_type(2)));
typedef float v8f __attribute__((ext_vector_type(8)));

// ---------------------------------------------------------------------------
// Kernel 1: hidden = ALPHA * init_x   (vectorized float4)
// ---------------------------------------------------------------------------
__global__ void gcnii_init(const float4* __restrict__ init_x,
                           float4* __restrict__ hidden, int n4) {
    int i = blockIdx.x * blockDim.x + threadIdx.x;
    if (i < n4) {
        float4 v = init_x[i];
        float4 o;
        o.x = ALPHA * v.x; o.y = ALPHA * v.y;
        o.z = ALPHA * v.z; o.w = ALPHA * v.w;
        hidden[i] = o;
    }
}

// ---------------------------------------------------------------------------
// Kernel 2: SpMM scatter. 16 threads per edge, 4 features each (float4 gather,
// 4x GLOBAL_ATOMIC_ADD_F32 scatter). (1-ALPHA) folded into the edge weight.
// ---------------------------------------------------------------------------
__global__ void gcnii_spmm(const float* __restrict__ x,
                           const float* __restrict__ ew,
                           const int* __restrict__ ei,
                           float* __restrict__ hidden) {
    int gid = blockIdx.x * blockDim.x + threadIdx.x;
    int e = gid >> 4;
    if (e >= NE) return;
    int f = (gid & 15) << 2;
    int src = ei[e];
    int dst = ei[NE + e];
    float w = (1.0f - ALPHA) * ew[e];
    float4 xv = *(const float4*)(x + src * DD + f);
    float* hp = hidden + dst * DD + f;
    atomicAdd(hp + 0, w * xv.x);
    atomicAdd(hp + 1, w * xv.y);
    atomicAdd(hp + 2, w * xv.z);
    atomicAdd(hp + 3, w * xv.w);
}

// ---------------------------------------------------------------------------
// Kernel 3: out = BETA*(hidden @ W^T + b) + (1-BETA)*hidden
// One wave (wave32) computes a 16-row strip x all 64 cols with
// V_WMMA_F32_16X16X4_F32 (full fp32). W^T staged in LDS in K-pair-interleaved
// layout: Wp[kp*KP + col*2 + par] = W[col][2*kp + par], so each lane's B
// operand {W^T[kk][col], W^T[kk+1][col]} is one contiguous ds_load_b64
// (no VGPR repacking movs, conflict-free banks via KP=160 pad).
//
// A (16x4 f32, 2 VGPRs): lanes 0-15 -> M=lane,    K={k0,k0+1};
//                        lanes 16-31 -> M=lane-16, K={k0+2,k0+3}.
// B (4x16 f32, 2 VGPRs): lanes 0-15 -> N=lane,    K={k0,k0+1};
//                        lanes 16-31 -> N=lane-16, K={k0+2,k0+3}.
// C/D (16x16 f32, 8 VGPRs): VGPR r -> rows r (lanes 0-15), r+8 (lanes 16-31).
// ---------------------------------------------------------------------------
__global__ __launch_bounds__(256) void gcnii_gemm(const float* __restrict__ hidden,
                                                  const float* __restrict__ W,
                                                  const float* __restrict__ b,
                                                  float* __restrict__ out) {
    __shared__ float Wp[(DD / 2) * KP];   // 32 * 160 * 4B = 20.5 KB

    int tid = threadIdx.x;
    // LDS-write-linear fill (conflict-free stores; W is 16 KB, read once).
    for (int i = tid; i < (DD / 2) * DD * 2; i += 256) {
        int kp  = i >> 7;        // K pair index  (0..31)
        int rem = i & 127;
        int col = rem >> 1;      // output column (0..63)
        int par = rem & 1;       // which K of the pair
        Wp[kp * KP + col * 2 + par] = W[col * DD + 2 * kp + par];
    }
    __syncthreads();

    int wave   = tid >> 5;
    int lane   = tid & 31;
    int lane16 = lane & 15;
    int khalf  = lane >> 4;           // 0: K pair {k0,k0+1}; 1: K pair {k0+2,k0+3}

    int rowTile = blockIdx.x * WAVES_PER_BLOCK + wave;
    if (rowTile >= ROW_TILES) return; // wave-uniform: EXEC stays all-ones for WMMA
    int rowBase = rowTile * 16;

    v8f acc[4];
    #pragma unroll
    for (int cj = 0; cj < 4; ++cj)
        acc[cj] = (v8f){0.f, 0.f, 0.f, 0.f, 0.f, 0.f, 0.f, 0.f};

    const float* arow = hidden + (size_t)(rowBase + lane16) * DD;

    #pragma unroll
    for (int k0 = 0; k0 < DD; k0 += 4) {
        int kk = k0 + 2 * khalf;
        v2f a = *(const v2f*)(arow + kk);                   // contiguous 8B load
        const float* bbase = &Wp[(k0 / 2 + khalf) * KP + lane16 * 2];
        #pragma unroll
        for (int cj = 0; cj < 4; ++cj) {
            v2f bm = *(const v2f*)(bbase + cj * 32);        // one ds_load_b64
            acc[cj] = __builtin_amdgcn_wmma_f32_16x16x4_f32(
                /*neg_a=*/false, a, /*neg_b=*/false, bm,
                /*c_mod=*/(short)0, acc[cj],
                /*reuse_a=*/false, /*reuse_b=*/false);
        }
    }

    // Fused epilogue: h = BETA*(acc + b) + (1-BETA)*hidden
    #pragma unroll
    for (int cj = 0; cj < 4; ++cj) {
        int col = cj * 16 + lane16;
        float bias = b[col];
        #pragma unroll
        for (int r = 0; r < 8; ++r) {
            int row = rowBase + r + 8 * khalf;
            float hid = hidden[(size_t)row * DD + col];
            out[(size_t)row * DD + col] =
                BETA * (acc[cj][r] + bias) + (1.0f - BETA) * hid;
        }
    }
}

// ---------------------------------------------------------------------------
extern "C" void kernel_launch(void* const* d_in, const int* in_sizes, int n_in,
                              void* d_out, int out_size, void* d_ws, size_t ws_size,
                              hipStream_t stream) {
    const float* x      = (const float*)d_in[0];
    const float* init_x = (const float*)d_in[1];
    const float* ew     = (const float*)d_in[2];
    const float* W      = (const float*)d_in[3];
    const float* b      = (const float*)d_in[4];
    const int*   ei     = (const int*)d_in[5];
    float*       out    = (float*)d_out;

    // hidden accumulator: prefer workspace; deterministic in-place fallback
    // (safe: gemm rows are partitioned per wave, reads precede writes).
    size_t hid_bytes = (size_t)NN * DD * sizeof(float);
    float* hidden = (ws_size >= hid_bytes) ? (float*)d_ws : out;

    int n4 = NN * DD / 4;
    gcnii_init<<<(n4 + 255) / 256, 256, 0, stream>>>(
        (const float4*)init_x, (float4*)hidden, n4);

    int spmm_threads = NE * 16;
    gcnii_spmm<<<(spmm_threads + 255) / 256, 256, 0, stream>>>(x, ew, ei, hidden);

    int gemm_blocks = (ROW_TILES + WAVES_PER_BLOCK - 1) / WAVES_PER_BLOCK;
    gcnii_gemm<<<gemm_blocks, 256, 0, stream>>>(hidden, W, b, out);
}